// RBFMetricHandler_28097676050903
// MI455X (gfx1250) — compile-verified
//
#include <hip/hip_runtime.h>
#include <math.h>

// ---------------------------------------------------------------------------
// RBF diagonal metric: out[b,d] = 1 / (sum_k exp(-0.5*dist[b,k]) * softplus(W)[k,d] + eps)
// dist[b,k] = sum_d bw[k,d]*(x[b,d]-c[k,d])^2
//           = sum_d bw*x^2  - 2*sum_d (bw*c)*x  + sum_d bw*c^2
// -> two f32 WMMA GEMMs (V_WMMA_F32_16X16X4_F32) with 2x2 register blocking,
//    fused quadratic+linear K-passes, fused exp / reciprocal epilogues.
// ---------------------------------------------------------------------------

typedef __attribute__((ext_vector_type(2))) float v2f;
typedef __attribute__((ext_vector_type(8))) float v8f;

#define B_ROWS 2048
#define K_CENT 256
#define D_DIM  512
#define KBIG   1024   // 2*D : [bw | -2*bw*c] concatenated K-dim of B^T for GEMM1
#define EPS    1e-3f

static __device__ __forceinline__ v8f wmma_f32(v2f a, v2f b, v8f c) {
    return __builtin_amdgcn_wmma_f32_16x16x4_f32(
        false, a, false, b, (short)0, c, false, false);
}

// ---------------------------------------------------------------------------
// Prep 1: elementwise over (k,d):
//   Bt[k][d]       = bw[k][d]                (B^T for GEMM1, quadratic half)
//   Bt[k][512+d]   = -2*bw[k][d]*c[k][d]     (B^T for GEMM1, linear half)
//   Wt[d][k]       = softplus(rw[k][d])      (B^T for GEMM2)
// ---------------------------------------------------------------------------
__global__ void __launch_bounds__(256)
prep_elem(const float* __restrict__ centers,
          const float* __restrict__ bw,
          const float* __restrict__ rw,
          float* __restrict__ Bt,
          float* __restrict__ Wt)
{
    int idx = blockIdx.x * 256 + threadIdx.x;       // over K*D = 131072
    int k = idx / D_DIM;
    int d = idx - k * D_DIM;
    float b = bw[idx];
    float c = centers[idx];
    float r = rw[idx];
    Bt[k * KBIG + d]         = b;
    Bt[k * KBIG + D_DIM + d] = -2.0f * b * c;
    // numerically stable softplus
    float sp = (r > 20.0f) ? r : log1pf(__expf(r));
    Wt[d * K_CENT + k] = sp;
}

// ---------------------------------------------------------------------------
// Prep 2: c2[k] = sum_d bw[k,d]*c[k,d]^2   (one block per k; deterministic)
// ---------------------------------------------------------------------------
__global__ void __launch_bounds__(256)
prep_c2(const float* __restrict__ centers,
        const float* __restrict__ bw,
        float* __restrict__ c2)
{
    int k = blockIdx.x;
    float s = 0.0f;
    for (int d = threadIdx.x; d < D_DIM; d += 256) {
        float c = centers[k * D_DIM + d];
        s += bw[k * D_DIM + d] * c * c;
    }
    __shared__ float red[256];
    red[threadIdx.x] = s;
    __syncthreads();
    for (int off = 128; off > 0; off >>= 1) {
        if (threadIdx.x < off) red[threadIdx.x] += red[threadIdx.x + off];
        __syncthreads();
    }
    if (threadIdx.x == 0) c2[k] = red[0];
}

// ---------------------------------------------------------------------------
// GEMM1 + exp epilogue: rbf[b,k] = exp(-0.5*(dist))
// Each wave computes a 32x32 output block = 2x2 WMMA tiles. The quadratic
// (x^2 * bw) and linear (x * -2*bw*c) contributions are fused in one K-pass:
// each x fragment is loaded once and feeds two wmmas (squared + raw), so a
// k-step of 4 is 6 loads -> 8 wmmas into 4 shared accumulators.
// ---------------------------------------------------------------------------
__global__ void __launch_bounds__(256)
gemm1_rbf(const float* __restrict__ x,
          const float* __restrict__ Bt,
          const float* __restrict__ c2,
          float* __restrict__ rbf)
{
    int wid  = blockIdx.x * 8 + (threadIdx.x >> 5);  // 512 waves total
    int lane = threadIdx.x & 31;
    int tileM = wid >> 3;        // 0..63  (M blocks of 32)
    int tileN = wid & 7;         // 0..7   (N blocks of 32)
    int m0 = tileM * 32;
    int n0 = tileN * 32;
    int half = lane >> 4;        // 0: K={0,1}, 1: K={2,3}
    int r    = lane & 15;

    const float* __restrict__ arow0 = x  + (size_t)(m0 + r) * D_DIM;
    const float* __restrict__ arow1 = arow0 + 16 * D_DIM;
    const float* __restrict__ brow0 = Bt + (size_t)(n0 + r) * KBIG;
    const float* __restrict__ brow1 = brow0 + 16 * KBIG;

    v8f acc00 = {}, acc01 = {}, acc10 = {}, acc11 = {};

    #pragma unroll 4
    for (int kk = 0; kk < D_DIM; kk += 4) {
        int idx = kk + 2 * half;
        v2f a0  = *(const v2f*)(arow0 + idx);           // x fragments (loaded once)
        v2f a1  = *(const v2f*)(arow1 + idx);
        v2f bq0 = *(const v2f*)(brow0 + idx);           // bw        (quadratic)
        v2f bq1 = *(const v2f*)(brow1 + idx);
        v2f bl0 = *(const v2f*)(brow0 + D_DIM + idx);   // -2*bw*c   (linear)
        v2f bl1 = *(const v2f*)(brow1 + D_DIM + idx);
        v2f a0s = a0 * a0;
        v2f a1s = a1 * a1;
        // interleave so no accumulator is used back-to-back (C->C dist 4)
        acc00 = wmma_f32(a0s, bq0, acc00);
        acc01 = wmma_f32(a0s, bq1, acc01);
        acc10 = wmma_f32(a1s, bq0, acc10);
        acc11 = wmma_f32(a1s, bq1, acc11);
        acc00 = wmma_f32(a0,  bl0, acc00);
        acc01 = wmma_f32(a0,  bl1, acc01);
        acc10 = wmma_f32(a1,  bl0, acc10);
        acc11 = wmma_f32(a1,  bl1, acc11);
    }

    float c2v0 = c2[n0 + r];
    float c2v1 = c2[n0 + 16 + r];
    // D layout: acc[i] = D[msub + i + 8*half][nsub + r]
    #pragma unroll
    for (int i = 0; i < 8; ++i) {
        int mA = m0 + i + 8 * half;
        int mB = mA + 16;
        rbf[(size_t)mA * K_CENT + n0 + r]      = __expf(-0.5f * (acc00[i] + c2v0));
        rbf[(size_t)mA * K_CENT + n0 + 16 + r] = __expf(-0.5f * (acc01[i] + c2v1));
        rbf[(size_t)mB * K_CENT + n0 + r]      = __expf(-0.5f * (acc10[i] + c2v0));
        rbf[(size_t)mB * K_CENT + n0 + 16 + r] = __expf(-0.5f * (acc11[i] + c2v1));
    }
}

// ---------------------------------------------------------------------------
// GEMM2 + reciprocal epilogue: out[b,d] = 1/(rbf x W + eps)
// 2x2 register blocking; K = 256.
// ---------------------------------------------------------------------------
__global__ void __launch_bounds__(256)
gemm2_out(const float* __restrict__ rbf,
          const float* __restrict__ Wt,
          float* __restrict__ out)
{
    int wid  = blockIdx.x * 8 + (threadIdx.x >> 5);  // 1024 waves total
    int lane = threadIdx.x & 31;
    int tileM = wid >> 4;        // 0..63  (M blocks of 32)
    int tileN = wid & 15;        // 0..15  (N blocks of 32)
    int m0 = tileM * 32;
    int n0 = tileN * 32;
    int half = lane >> 4;
    int r    = lane & 15;

    const float* __restrict__ arow0 = rbf + (size_t)(m0 + r) * K_CENT;
    const float* __restrict__ arow1 = arow0 + 16 * K_CENT;
    const float* __restrict__ brow0 = Wt  + (size_t)(n0 + r) * K_CENT;
    const float* __restrict__ brow1 = brow0 + 16 * K_CENT;

    v8f acc00 = {}, acc01 = {}, acc10 = {}, acc11 = {};

    #pragma unroll 4
    for (int kk = 0; kk < K_CENT; kk += 4) {
        int idx = kk + 2 * half;
        v2f a0 = *(const v2f*)(arow0 + idx);
        v2f a1 = *(const v2f*)(arow1 + idx);
        v2f b0 = *(const v2f*)(brow0 + idx);
        v2f b1 = *(const v2f*)(brow1 + idx);
        acc00 = wmma_f32(a0, b0, acc00);
        acc01 = wmma_f32(a0, b1, acc01);
        acc10 = wmma_f32(a1, b0, acc10);
        acc11 = wmma_f32(a1, b1, acc11);
    }

    #pragma unroll
    for (int i = 0; i < 8; ++i) {
        int mA = m0 + i + 8 * half;
        int mB = mA + 16;
        out[(size_t)mA * D_DIM + n0 + r]      = 1.0f / (acc00[i] + EPS);
        out[(size_t)mA * D_DIM + n0 + 16 + r] = 1.0f / (acc01[i] + EPS);
        out[(size_t)mB * D_DIM + n0 + r]      = 1.0f / (acc10[i] + EPS);
        out[(size_t)mB * D_DIM + n0 + 16 + r] = 1.0f / (acc11[i] + EPS);
    }
}

// ---------------------------------------------------------------------------
// Launch
// ---------------------------------------------------------------------------
extern "C" void kernel_launch(void* const* d_in, const int* in_sizes, int n_in,
                              void* d_out, int out_size, void* d_ws, size_t ws_size,
                              hipStream_t stream) {
    const float* x       = (const float*)d_in[0];  // (2048, 512)
    const float* centers = (const float*)d_in[1];  // (256, 512)
    const float* bw      = (const float*)d_in[2];  // (256, 512)
    const float* rw      = (const float*)d_in[3];  // (256, 512)
    float* out = (float*)d_out;                    // (2048, 512)

    float* ws  = (float*)d_ws;
    float* Bt  = ws;                               // 256*1024 = 262144 floats
    float* Wt  = Bt + (size_t)K_CENT * KBIG;       // 512*256  = 131072 floats
    float* c2  = Wt + (size_t)D_DIM * K_CENT;      // 256 floats
    float* rbf = c2 + K_CENT;                      // 2048*256 = 524288 floats

    prep_elem<<<(K_CENT * D_DIM) / 256, 256, 0, stream>>>(centers, bw, rw, Bt, Wt);
    prep_c2<<<K_CENT, 256, 0, stream>>>(centers, bw, c2);
    // GEMM1: 64 x 8 macro-tiles of 32x32, 8 waves/block -> 64 blocks
    gemm1_rbf<<<(B_ROWS / 32) * (K_CENT / 32) / 8, 256, 0, stream>>>(x, Bt, c2, rbf);
    // GEMM2: 64 x 16 macro-tiles of 32x32, 8 waves/block -> 128 blocks
    gemm2_out<<<(B_ROWS / 32) * (D_DIM / 32) / 8, 256, 0, stream>>>(rbf, Wt, out);
}